// GCN_85298050499264
// MI455X (gfx1250) — compile-verified
//
#include <hip/hip_runtime.h>
#include <stdint.h>

typedef __attribute__((ext_vector_type(2))) float v2f;
typedef __attribute__((ext_vector_type(8))) float v8f;

#define NNODES 50000
#define DIN    128
#define NEDGES 800000
#define NPAD   512   // 500 padded up to a multiple of 64

// ---------------------------------------------------------------------------
// Degree / normalization kernels
// ---------------------------------------------------------------------------
__global__ void fill_deg_kernel(float* deg, int n) {
    int i = blockIdx.x * blockDim.x + threadIdx.x;
    if (i < n) deg[i] = 1.0f;  // self-loop contributes 1 to every node's degree
}

__global__ void count_deg_kernel(const long long* dst, float* deg, int E) {
    int e = blockIdx.x * blockDim.x + threadIdx.x;
    if (e < E) atomicAdd(&deg[(int)dst[e]], 1.0f);
}

__global__ void dinv_kernel(float* deg, int n) {
    int i = blockIdx.x * blockDim.x + threadIdx.x;
    if (i < n) {
        float d = deg[i];
        deg[i] = (d > 0.0f) ? rsqrtf(d) : 0.0f;  // deg >= 1 with self-loops
    }
}

// ---------------------------------------------------------------------------
// Weight / bias padding: W[K,N] -> Wp[K,Npad] zero-filled, b[N] -> bp[Npad]
// ---------------------------------------------------------------------------
__global__ void pad_weight_kernel(const float* __restrict__ W, float* __restrict__ Wp,
                                  int K, int N, int Npad) {
    int i = blockIdx.x * blockDim.x + threadIdx.x;
    if (i < K * Npad) {
        int r = i / Npad, c = i % Npad;
        Wp[i] = (c < N) ? W[(size_t)r * N + c] : 0.0f;
    }
}

__global__ void pad_bias_kernel(const float* __restrict__ b, float* __restrict__ bp,
                                int N, int Npad) {
    int i = blockIdx.x * blockDim.x + threadIdx.x;
    if (i < Npad) bp[i] = (i < N) ? b[i] : 0.0f;
}

// ---------------------------------------------------------------------------
// WMMA fp32 GEMM: C[M, Npad] = A[M, K(lda)] @ B[K, Npad(ldb)] (+bias, +relu).
// All dimensions pre-padded: M % 16 == 0, Npad % 64 == 0, K % 4 == 0, so
// every load/store is unconditionally in bounds (no exec-mask churn) and
// EXEC is all-1s at every WMMA. One wave owns a 16x64 strip: one A fragment
// (global_load_b64) feeds 4 independent v_wmma_f32_16x16x4_f32 per k-step.
//
// Fragment layouts (ISA 7.12.2, wave32):
//   A 16x4 : lanes 0-15 -> M=lane, VGPR0=K0, VGPR1=K1; lanes 16-31 -> K2,K3
//   B 4x16 : VGPR0 -> rows K0 (lanes 0-15) / K2 (lanes 16-31); VGPR1 -> K1/K3
//   C 16x16: VGPR r -> M = r + 8*(lane>=16), N = lane&15
// ---------------------------------------------------------------------------
__global__ void gemm_wmma_f32(const float* __restrict__ A,
                              const float* __restrict__ B,
                              const float* __restrict__ bias,
                              float* __restrict__ C,
                              int M, int Npad, int K,
                              int lda, int ldb, int ldc, int doRelu) {
    const int wavesPerBlock = blockDim.x >> 5;
    const int waveId = blockIdx.x * wavesPerBlock + (threadIdx.x >> 5);
    const int lane   = threadIdx.x & 31;

    const int ngroups = Npad >> 6;          // 16x64 strips per row band
    const int mtiles  = M >> 4;
    if (waveId >= mtiles * ngroups) return; // wave-uniform exit

    const int mt = waveId / ngroups;
    const int ng = waveId % ngroups;

    const int row   = mt * 16 + (lane & 15);
    const int col0  = ng * 64 + (lane & 15);
    const int khalf = (lane >> 4) << 1;     // 0 (lanes 0-15) or 2 (lanes 16-31)

    const float* Arow = A + (size_t)row * lda + khalf;   // even -> 8B aligned
    const float* Bcol = B + (size_t)khalf * ldb + col0;

    v8f acc0 = {}, acc1 = {}, acc2 = {}, acc3 = {};
    for (int k = 0; k < K; k += 4) {
        v2f a = *(const v2f*)(Arow + k);
        const float* Bk = Bcol + (size_t)k * ldb;
        v2f b0, b1, b2, b3;
        b0.x = Bk[0];  b0.y = Bk[ldb];
        b1.x = Bk[16]; b1.y = Bk[ldb + 16];
        b2.x = Bk[32]; b2.y = Bk[ldb + 32];
        b3.x = Bk[48]; b3.y = Bk[ldb + 48];
        acc0 = __builtin_amdgcn_wmma_f32_16x16x4_f32(false, a, false, b0, (short)0, acc0, false, false);
        acc1 = __builtin_amdgcn_wmma_f32_16x16x4_f32(false, a, false, b1, (short)0, acc1, false, false);
        acc2 = __builtin_amdgcn_wmma_f32_16x16x4_f32(false, a, false, b2, (short)0, acc2, false, false);
        acc3 = __builtin_amdgcn_wmma_f32_16x16x4_f32(false, a, false, b3, (short)0, acc3, false, false);
    }

    const int rbase = mt * 16 + (lane >> 4) * 8;
    float* Cp = C + (size_t)rbase * ldc + col0;
    v8f accs[4] = {acc0, acc1, acc2, acc3};
    #pragma unroll
    for (int t = 0; t < 4; ++t) {
        const float bv = bias ? bias[col0 + 16 * t] : 0.0f;
        #pragma unroll
        for (int r = 0; r < 8; ++r) {
            float v = accs[t][r] + bv;
            if (doRelu) v = fmaxf(v, 0.0f);
            Cp[(size_t)r * ldc + 16 * t] = v;
        }
    }
}

// ---------------------------------------------------------------------------
// GCN aggregation
// ---------------------------------------------------------------------------
// acc[n,f] = xw[n,f] * dinv[n]^2  (self-loop term; also initializes acc)
__global__ void self_loop_kernel(const float* __restrict__ xw,
                                 const float* __restrict__ dinv,
                                 float* __restrict__ acc, int n) {
    int i = blockIdx.x * blockDim.x + threadIdx.x;  // n = NNODES*DIN
    if (i < n) {
        float w = dinv[i >> 7];  // DIN == 128
        acc[i] = xw[i] * w * w;
    }
}

// One wave per edge; 128 features / 32 lanes = 4 atomic adds per lane.
__global__ void scatter_kernel(const long long* __restrict__ src,
                               const long long* __restrict__ dst,
                               const float* __restrict__ xw,
                               const float* __restrict__ dinv,
                               float* __restrict__ acc, int E) {
    int tid  = blockIdx.x * blockDim.x + threadIdx.x;
    int e    = tid >> 5;
    int lane = tid & 31;
    if (e >= E) return;
    int s = (int)src[e];
    int d = (int)dst[e];
    float w = dinv[s] * dinv[d];
    const float* xrow = xw  + (size_t)s * DIN;
    float*       arow = acc + (size_t)d * DIN;
    #pragma unroll
    for (int f = lane; f < DIN; f += 32)
        atomicAdd(&arow[f], xrow[f] * w);
}

// h0 = relu(acc + conv_b), in place (DIN == 128 so feature = i & 127)
__global__ void bias_relu_kernel(float* __restrict__ acc,
                                 const float* __restrict__ b, int n) {
    int i = blockIdx.x * blockDim.x + threadIdx.x;
    if (i < n) acc[i] = fmaxf(acc[i] + b[i & 127], 0.0f);
}

// ---------------------------------------------------------------------------
// Final 500 -> 1 projection: one wave per node, shuffle reduction (wave32)
// ---------------------------------------------------------------------------
__global__ void final_dot_kernel(const float* __restrict__ h,
                                 const float* __restrict__ Wo,
                                 const float* __restrict__ bo,
                                 float* __restrict__ out, int M, int K, int ldh) {
    const int wavesPerBlock = blockDim.x >> 5;
    const int node = blockIdx.x * wavesPerBlock + (threadIdx.x >> 5);
    const int lane = threadIdx.x & 31;
    if (node >= M) return;
    const float* row = h + (size_t)node * ldh;
    float s = 0.0f;
    for (int k = lane; k < K; k += 32) s += row[k] * Wo[k];
    #pragma unroll
    for (int off = 16; off > 0; off >>= 1) s += __shfl_xor(s, off, 32);
    if (lane == 0) out[node] = s + bo[0];
}

// ---------------------------------------------------------------------------
// Launch
// ---------------------------------------------------------------------------
extern "C" void kernel_launch(void* const* d_in, const int* in_sizes, int n_in,
                              void* d_out, int out_size, void* d_ws, size_t ws_size,
                              hipStream_t stream) {
    const float*     x      = (const float*)d_in[0];
    const long long* eidx   = (const long long*)d_in[1];  // int64 [2, 800000]
    const float*     conv_W = (const float*)d_in[2];
    const float*     conv_b = (const float*)d_in[3];
    const float*     W1     = (const float*)d_in[4];
    const float*     b1     = (const float*)d_in[5];
    const float*     W2     = (const float*)d_in[6];
    const float*     b2     = (const float*)d_in[7];
    const float*     W3     = (const float*)d_in[8];
    const float*     b3     = (const float*)d_in[9];
    const float*     Wo     = (const float*)d_in[10];
    const float*     bo     = (const float*)d_in[11];
    float* out = (float*)d_out;

    const long long* e_src = eidx;            // edge_index[0]
    const long long* e_dst = eidx + NEDGES;   // edge_index[1]

    // Workspace layout (floats):
    //   xw  : NNODES*128          (25.6 MB)
    //   deg : NNODES              (0.2 MB)   -> becomes dinv in place
    //   acc : NNODES*128          (25.6 MB)  -> becomes h0 in place
    //   hA  : NNODES*NPAD         (102.4 MB) h1, then h3
    //   hB  : NNODES*NPAD         (102.4 MB) h2
    //   W1p : 128*NPAD, W2p/W3p : 500*NPAD, b1p/b2p/b3p : NPAD
    float* ws  = (float*)d_ws;
    float* xw  = ws;
    float* deg = xw  + (size_t)NNODES * DIN;
    float* acc = deg + NNODES;
    float* hA  = acc + (size_t)NNODES * DIN;
    float* hB  = hA  + (size_t)NNODES * NPAD;
    float* W1p = hB  + (size_t)NNODES * NPAD;
    float* W2p = W1p + (size_t)DIN * NPAD;
    float* W3p = W2p + (size_t)500 * NPAD;
    float* b1p = W3p + (size_t)500 * NPAD;
    float* b2p = b1p + NPAD;
    float* b3p = b2p + NPAD;

    const int BLK = 256;
    const int wavesPerBlock = BLK / 32;

    // 0) pad weights/biases once per call (tiny, fully parallel)
    pad_weight_kernel<<<((DIN * NPAD) + BLK - 1) / BLK, BLK, 0, stream>>>(W1, W1p, DIN, 500, NPAD);
    pad_weight_kernel<<<((500 * NPAD) + BLK - 1) / BLK, BLK, 0, stream>>>(W2, W2p, 500, 500, NPAD);
    pad_weight_kernel<<<((500 * NPAD) + BLK - 1) / BLK, BLK, 0, stream>>>(W3, W3p, 500, 500, NPAD);
    pad_bias_kernel<<<1, NPAD, 0, stream>>>(b1, b1p, 500, NPAD);
    pad_bias_kernel<<<1, NPAD, 0, stream>>>(b2, b2p, 500, NPAD);
    pad_bias_kernel<<<1, NPAD, 0, stream>>>(b3, b3p, 500, NPAD);

    // 1) degree with self-loops, then dinv = rsqrt(deg) in place
    fill_deg_kernel<<<(NNODES + BLK - 1) / BLK, BLK, 0, stream>>>(deg, NNODES);
    count_deg_kernel<<<(NEDGES + BLK - 1) / BLK, BLK, 0, stream>>>(e_dst, deg, NEDGES);
    dinv_kernel<<<(NNODES + BLK - 1) / BLK, BLK, 0, stream>>>(deg, NNODES);

    // 2) xw = x @ conv_W   (M=50000, N=128, K=128; already multiple of 64)
    {
        int waves = (NNODES / 16) * (DIN / 64);
        gemm_wmma_f32<<<(waves + wavesPerBlock - 1) / wavesPerBlock, BLK, 0, stream>>>(
            x, conv_W, nullptr, xw, NNODES, DIN, DIN, DIN, DIN, DIN, 0);
    }

    // 3) aggregation: self-loop init + per-edge atomic scatter
    self_loop_kernel<<<((NNODES * DIN) + BLK - 1) / BLK, BLK, 0, stream>>>(
        xw, deg, acc, NNODES * DIN);
    {
        long long threads = (long long)NEDGES * 32;
        scatter_kernel<<<(int)((threads + BLK - 1) / BLK), BLK, 0, stream>>>(
            e_src, e_dst, xw, deg, acc, NEDGES);
    }

    // 4) h0 = relu(acc + conv_b), in place
    bias_relu_kernel<<<((NNODES * DIN) + BLK - 1) / BLK, BLK, 0, stream>>>(
        acc, conv_b, NNODES * DIN);

    // 5) MLP stack, bias+relu fused into the WMMA epilogue; N padded to 512
    {
        int waves  = (NNODES / 16) * (NPAD / 64);
        int blocks = (waves + wavesPerBlock - 1) / wavesPerBlock;
        // h1 = relu(h0 @ W1 + b1)   K=128, lda=128
        gemm_wmma_f32<<<blocks, BLK, 0, stream>>>(acc, W1p, b1p, hA,
                                                  NNODES, NPAD, DIN, DIN, NPAD, NPAD, 1);
        // h2 = relu(h1 @ W2 + b2)   K=500, lda=512
        gemm_wmma_f32<<<blocks, BLK, 0, stream>>>(hA, W2p, b2p, hB,
                                                  NNODES, NPAD, 500, NPAD, NPAD, NPAD, 1);
        // h3 = relu(h2 @ W3 + b3)   K=500 (reuse hA)
        gemm_wmma_f32<<<blocks, BLK, 0, stream>>>(hB, W3p, b3p, hA,
                                                  NNODES, NPAD, 500, NPAD, NPAD, NPAD, 1);
    }

    // 6) out = h3 @ Wo + bo  (500 -> 1), wave-per-node reduction
    {
        int waves = NNODES;
        final_dot_kernel<<<(waves + wavesPerBlock - 1) / wavesPerBlock, BLK, 0, stream>>>(
            hA, Wo, bo, out, NNODES, 500, NPAD);
    }
}